// MaskPooling_83056077570584
// MI455X (gfx1250) — compile-verified
//
#include <hip/hip_runtime.h>

typedef float f4  __attribute__((ext_vector_type(4)));
typedef float v2f __attribute__((ext_vector_type(2)));
typedef float v8f __attribute__((ext_vector_type(8)));

#define HB   16
#define HC   256
#define HH   128
#define HWW  128
#define MHH  32
#define MWW  32
#define PLANE (HH * HWW)            // 16384 elements per (b,c) plane
#define NMASK (HB * PLANE)          // 262144 mask samples
#define WS_MASK_OFF 4096            // byte offset of u8 mask buffer in ws
// ws layout: [0..255] sum_all, [256..511] sum_ch, [512] cnt_ch, then u8 mask at +4096

// ---------------------------------------------------------------------------
// Kernel 1: bilinear-upsample (half-pixel centers, edge clamp) + threshold.
// Stores 0/1 bytes, block-reduces the count, atomics into cnt.
// ---------------------------------------------------------------------------
__global__ __launch_bounds__(256) void mask_build_kernel(const float* __restrict__ mask,
                                                         unsigned char* __restrict__ mout,
                                                         float* __restrict__ cnt)
{
    const int idx = blockIdx.x * 256 + threadIdx.x;   // 0..262143
    const int b   = idx >> 14;
    const int rem = idx & 16383;
    const int h   = rem >> 7;
    const int w   = rem & 127;

    // src coord = (dst + 0.5) * (in/out) - 0.5 ; in/out = 32/128 = 0.25
    const float fy  = (h + 0.5f) * 0.25f - 0.5f;
    const float fx  = (w + 0.5f) * 0.25f - 0.5f;
    const float fy0 = floorf(fy);
    const float fx0 = floorf(fx);
    const float ty  = fy - fy0;
    const float tx  = fx - fx0;
    const int y0 = (int)fy0, x0 = (int)fx0;
    const int y0c = min(max(y0,     0), MHH - 1);
    const int y1c = min(max(y0 + 1, 0), MHH - 1);
    const int x0c = min(max(x0,     0), MWW - 1);
    const int x1c = min(max(x0 + 1, 0), MWW - 1);

    const float* mb = mask + b * (MHH * MWW);
    const float v00 = mb[y0c * MWW + x0c];
    const float v01 = mb[y0c * MWW + x1c];
    const float v10 = mb[y1c * MWW + x0c];
    const float v11 = mb[y1c * MWW + x1c];
    const float v0  = v00 + tx * (v01 - v00);
    const float v1  = v10 + tx * (v11 - v10);
    const float v   = v0  + ty * (v1  - v0);
    const float m   = (v > 0.5f) ? 1.0f : 0.0f;

    mout[idx] = (unsigned char)m;

    __shared__ float red[256];
    red[threadIdx.x] = m;
    __syncthreads();
    for (int s = 128; s > 0; s >>= 1) {
        if (threadIdx.x < s) red[threadIdx.x] += red[threadIdx.x + s];
        __syncthreads();
    }
    if (threadIdx.x == 0) atomicAdd(cnt, red[0]);
}

// ---------------------------------------------------------------------------
// Kernel 2: one block per (b,c) plane. Streaming B128 non-temporal loads of x,
// packed u8 mask loads, then a WMMA-based block reduction:
//   A = ones(16x4), B = partial sums  =>  D[m][n] = sum_k B[k][n]  (+C chain)
// 4 tiles of 64 values cover all 256 per-thread partials per stream.
// ---------------------------------------------------------------------------
__global__ __launch_bounds__(256) void plane_reduce_kernel(const float* __restrict__ x,
                                                           const unsigned char* __restrict__ mask,
                                                           float* __restrict__ sum_all,
                                                           float* __restrict__ sum_ch)
{
    const int plane = blockIdx.x;        // b*256 + c  (matches (B,C,H,W) layout)
    const int b = plane >> 8;
    const int c = plane & 255;
    const int t = threadIdx.x;

    const f4*           xp = (const f4*)(x + (size_t)plane * PLANE);
    const unsigned int* mp = (const unsigned int*)(mask + (size_t)b * PLANE);

    float s_all = 0.0f, s_ch = 0.0f;
#pragma unroll
    for (int i = 0; i < 16; ++i) {
        const f4 v           = __builtin_nontemporal_load(&xp[i * 256 + t]);
        const unsigned int q = mp[i * 256 + t];
        s_all += v.x + v.y + v.z + v.w;
        s_ch  += (q & 0x000000ffu) ? v.x : 0.0f;
        s_ch  += (q & 0x0000ff00u) ? v.y : 0.0f;
        s_ch  += (q & 0x00ff0000u) ? v.z : 0.0f;
        s_ch  += (q & 0xff000000u) ? v.w : 0.0f;
    }

    __shared__ float red[512];
    red[t]       = s_all;
    red[256 + t] = s_ch;
    __syncthreads();

    // WMMA reduction, executed redundantly by all 8 waves (EXEC all-ones each).
    const int lane = t & 31;
    v2f a; a.x = 1.0f; a.y = 1.0f;      // A = ones(16x4)
    v8f c_all = {}; v8f c_chm = {};
#pragma unroll
    for (int tile = 0; tile < 4; ++tile) {
        v2f b0, b1;
        b0.x = red[tile * 64 + lane];
        b0.y = red[tile * 64 + 32 + lane];
        b1.x = red[256 + tile * 64 + lane];
        b1.y = red[256 + tile * 64 + 32 + lane];
        c_all = __builtin_amdgcn_wmma_f32_16x16x4_f32(false, a, false, b0,
                                                      (short)0, c_all, false, false);
        c_chm = __builtin_amdgcn_wmma_f32_16x16x4_f32(false, a, false, b1,
                                                      (short)0, c_chm, false, false);
    }
    // c[0] row M=0: lane n<16 holds the column-n sum (lanes 16..31 replicate M=8).
    float ta = c_all[0];
    float tc = c_chm[0];
#pragma unroll
    for (int o = 8; o >= 1; o >>= 1) {   // xor masks stay within lanes 0..15
        ta += __shfl_xor(ta, o, 32);
        tc += __shfl_xor(tc, o, 32);
    }
    if (t == 0) {
        atomicAdd(&sum_all[c], ta);
        atomicAdd(&sum_ch[c], tc);
    }
}

// ---------------------------------------------------------------------------
// Kernel 3: finalize -> d_out = [unch(256) | ch(256)]
// ---------------------------------------------------------------------------
__global__ __launch_bounds__(256) void finalize_kernel(const float* __restrict__ acc,
                                                       float* __restrict__ out)
{
    const int c = threadIdx.x;
    const float sa   = acc[c];
    const float sc   = acc[256 + c];
    const float cntc = acc[512];
    const float cnta = (float)(HB * HH * HWW);   // 262144
    out[c]       = (sa - sc) / (cnta - cntc);    // unch
    out[256 + c] = sc / cntc;                    // ch
}

extern "C" void kernel_launch(void* const* d_in, const int* in_sizes, int n_in,
                              void* d_out, int out_size, void* d_ws, size_t ws_size,
                              hipStream_t stream)
{
    const float* x    = (const float*)d_in[0];   // (16,256,128,128) f32
    const float* mask = (const float*)d_in[1];   // (16,32,32) f32

    float*         acc  = (float*)d_ws;                          // 513 floats used
    unsigned char* mbuf = (unsigned char*)d_ws + WS_MASK_OFF;    // 262144 bytes

    hipMemsetAsync(d_ws, 0, WS_MASK_OFF, stream);  // zero accumulators (capture-safe)

    mask_build_kernel<<<NMASK / 256, 256, 0, stream>>>(mask, mbuf, acc + 512);
    plane_reduce_kernel<<<HB * HC, 256, 0, stream>>>(x, mbuf, acc, acc + 256);
    finalize_kernel<<<1, 256, 0, stream>>>(acc, (float*)d_out);
}